// STEQuantizer_2345052144226
// MI455X (gfx1250) — compile-verified
//
#include <hip/hip_runtime.h>
#include <math.h>

// out[n, d] = W[d, argmax_k x[n, k]]   (N=8192, K=4096, D=1024, all fp32)
//
// Memory-bound: 134MB x-stream (NT b128), W kept L2-resident (16.8MB of 192MB
// L2), 33.6MB NT output stream. ~8us roofline at 23.3 TB/s. No WMMA: a dense
// one-hot matmul would be 68.7 GFLOP of wasted work vs this gather.

typedef __attribute__((ext_vector_type(4))) float v4f;

#define TPB 256          // 8 wave32s per block
#define NWAVE (TPB / 32)

static constexpr int KDIM = 4096;   // quant dim (argmax axis)
static constexpr int DDIM = 1024;   // output dim

// CDNA5 async global->LDS path (ASYNCcnt-tracked DMA into LDS, no VGPR hop).
// clang declares the builtin with typed int* params: (global int*, lds int*,
// imm offset, imm cpol).
#if defined(__AMDGCN__) &&                                     \
    __has_builtin(__builtin_amdgcn_global_load_async_to_lds_b32) && \
    __has_builtin(__builtin_amdgcn_s_wait_asynccnt)
#define HAVE_ASYNC_LDS 1
typedef __attribute__((address_space(3))) int lds_int_t;
typedef __attribute__((address_space(1))) int glob_int_t;
#else
#define HAVE_ASYNC_LDS 0
#endif

// ---------------------------------------------------------------------------
// Block-wide argmax of one row of length K (first-occurrence tie-break).
// Nontemporal b128 loads: x is read exactly once; don't pollute L2 (keep W hot).
// ---------------------------------------------------------------------------
__device__ __forceinline__ int block_row_argmax(const float* __restrict__ xrow) {
    float bv = -INFINITY;
    int   bi = 0x7fffffff;

    const v4f* xr = reinterpret_cast<const v4f*>(xrow);
    constexpr int nchunk = KDIM >> 2;  // float4 chunks
#pragma unroll
    for (int c4 = threadIdx.x; c4 < nchunk; c4 += TPB) {
        v4f v = __builtin_nontemporal_load(xr + c4);  // global_load_b128 th:NT
        const int base = c4 << 2;
#pragma unroll
        for (int e = 0; e < 4; ++e) {
            const float f  = v[e];
            const int   ii = base + e;
            if (f > bv || (f == bv && ii < bi)) { bv = f; bi = ii; }
        }
    }

    // wave32 butterfly reduction (gfx1250 is wave32-only)
#pragma unroll
    for (int off = 16; off > 0; off >>= 1) {
        const float ov = __shfl_xor(bv, off, 32);
        const int   oi = __shfl_xor(bi, off, 32);
        if (ov > bv || (ov == bv && oi < bi)) { bv = ov; bi = oi; }
    }

    // cross-wave reduction through LDS
    __shared__ float s_v[NWAVE];
    __shared__ int   s_i[NWAVE];
    __shared__ int   s_c;
    const int lane = threadIdx.x & 31;
    const int wave = threadIdx.x >> 5;
    if (lane == 0) { s_v[wave] = bv; s_i[wave] = bi; }
    __syncthreads();
    if (threadIdx.x == 0) {
        float fv = s_v[0];
        int   fi = s_i[0];
#pragma unroll
        for (int w = 1; w < NWAVE; ++w) {
            const float ov = s_v[w];
            const int   oi = s_i[w];
            if (ov > fv || (ov == fv && oi < fi)) { fv = ov; fi = oi; }
        }
        s_c = fi;
    }
    __syncthreads();
    return s_c;
}

// ---------------------------------------------------------------------------
// W (D x K, row-major) -> Wt (K x D, row-major), LDS 32x33 tile transpose.
// CDNA5: fill the LDS tile with GLOBAL_LOAD_ASYNC_TO_LDS_B32 (memory->LDS DMA,
// ASYNCcnt-tracked, no VGPR round-trip), drain with s_wait_asynccnt.
// ---------------------------------------------------------------------------
__global__ __launch_bounds__(TPB) void transpose_w_kernel(
    const float* __restrict__ W, float* __restrict__ Wt) {
    __shared__ float tile[32][33];  // +1 pad: no LDS bank conflicts
    const int k0 = blockIdx.x * 32;
    const int d0 = blockIdx.y * 32;
    const int tx = threadIdx.x & 31;
    const int ty = threadIdx.x >> 5;  // 0..7

#if HAVE_ASYNC_LDS
#pragma unroll
    for (int i = 0; i < 4; ++i) {
        const int d = d0 + ty + i * 8;
        const float* gsrc = W + (size_t)d * KDIM + (k0 + tx);  // coalesced in k
        __builtin_amdgcn_global_load_async_to_lds_b32(
            (glob_int_t*)gsrc, (lds_int_t*)&tile[ty + i * 8][tx],
            /*offset=*/0, /*cpol=*/0);
    }
    __builtin_amdgcn_s_wait_asynccnt(0);   // this wave's LDS writes landed
#else
#pragma unroll
    for (int i = 0; i < 4; ++i) {
        const int d = d0 + ty + i * 8;
        tile[ty + i * 8][tx] = W[(size_t)d * KDIM + (k0 + tx)];
    }
#endif
    __syncthreads();                       // make all waves' tile data visible
#pragma unroll
    for (int i = 0; i < 4; ++i) {
        const int k = k0 + ty + i * 8;
        Wt[(size_t)k * DDIM + (d0 + tx)] = tile[tx][ty + i * 8]; // coalesced in d
    }
}

// ---------------------------------------------------------------------------
// Fast path: argmax + contiguous row copy out[n,:] = Wt[c,:] (L2-hot).
// ---------------------------------------------------------------------------
__global__ __launch_bounds__(TPB) void argmax_copy_kernel(
    const float* __restrict__ x, const float* __restrict__ Wt,
    float* __restrict__ out) {
    const int n = blockIdx.x;
    const int c = block_row_argmax(x + (size_t)n * KDIM);

    const v4f* src = reinterpret_cast<const v4f*>(Wt + (size_t)c * DDIM);
    v4f*       dst = reinterpret_cast<v4f*>(out + (size_t)n * DDIM);
#pragma unroll
    for (int i = threadIdx.x; i < (DDIM >> 2); i += TPB) {
        const v4f v = src[i];                    // RT load: reused across rows (L2)
        __builtin_nontemporal_store(v, dst + i); // write-once output, NT b128
    }
}

// ---------------------------------------------------------------------------
// Fallback (scratch too small for Wt): strided column gather, L2-backed.
// ---------------------------------------------------------------------------
__global__ __launch_bounds__(TPB) void argmax_gather_kernel(
    const float* __restrict__ x, const float* __restrict__ W,
    float* __restrict__ out) {
    const int n = blockIdx.x;
    const int c = block_row_argmax(x + (size_t)n * KDIM);

    float* orow = out + (size_t)n * DDIM;
#pragma unroll
    for (int d0 = threadIdx.x * 4; d0 < DDIM; d0 += TPB * 4) {
        v4f o;
#pragma unroll
        for (int j = 0; j < 4; ++j) {
            o[j] = W[(size_t)(d0 + j) * KDIM + c];  // W fits in L2 -> hits
        }
        __builtin_nontemporal_store(o, reinterpret_cast<v4f*>(orow + d0));
    }
}

// ---------------------------------------------------------------------------
extern "C" void kernel_launch(void* const* d_in, const int* in_sizes, int n_in,
                              void* d_out, int out_size, void* d_ws,
                              size_t ws_size, hipStream_t stream) {
    const float* x = (const float*)d_in[0];   // [N, K]
    const float* W = (const float*)d_in[1];   // [D, K]
    float*     out = (float*)d_out;           // [N, D]

    const int N = in_sizes[0] / KDIM;         // 8192

    const size_t wt_bytes = (size_t)KDIM * (size_t)DDIM * sizeof(float); // 16 MiB

    if (ws_size >= wt_bytes) {
        float* Wt = (float*)d_ws;
        transpose_w_kernel<<<dim3(KDIM / 32, DDIM / 32), TPB, 0, stream>>>(W, Wt);
        argmax_copy_kernel<<<N, TPB, 0, stream>>>(x, Wt, out);
    } else {
        argmax_gather_kernel<<<N, TPB, 0, stream>>>(x, W, out);
    }
}